// fashionLoss_30648886624644
// MI455X (gfx1250) — compile-verified
//
#include <hip/hip_runtime.h>
#include <stdint.h>

typedef float v2f __attribute__((ext_vector_type(2)));
typedef float v8f __attribute__((ext_vector_type(8)));
typedef int   v4i __attribute__((ext_vector_type(4)));

#define LOSS_EPS 1e-10f
#define BTOT     524288
#define TILE     256
#define NBLK     (BTOT / TILE)   // 2048
#define LOG2E    1.44269504088896340736f
#define LN2      0.69314718055994530942f

// ---- wave32 sum via V_WMMA_F32_16X16X4_F32 ----
// A: a[0]=v -> A[m][0]=v(lane m, m<16), A[m][2]=v(lane m+16); a[1]=0 -> K=1,3 zero.
// B: ones(4x16).  D[m][n] = v[m] + v[m+16] = s[m]  (all 16 columns identical).
// Lane n<16 holds s[0..7] in its 8 D regs, lane n>=16 holds s[8..15];
// local 8-add + one shfl_xor(16) completes the 32-lane sum. Requires full EXEC.
__device__ __forceinline__ float wave_sum_wmma(float v) {
    v2f a; a[0] = v;    a[1] = 0.0f;
    v2f b; b[0] = 1.0f; b[1] = 1.0f;
    v8f c = {0.f, 0.f, 0.f, 0.f, 0.f, 0.f, 0.f, 0.f};
    v8f d = __builtin_amdgcn_wmma_f32_16x16x4_f32(
        false, a, false, b, (short)0, c, false, false);
    float t = d[0] + d[1] + d[2] + d[3] + d[4] + d[5] + d[6] + d[7];
    t += __shfl_xor(t, 16, 32);
    return t;
}

// -sum_j log(softmax(x)_j + eps) * lbl_j, max-subtracted softmax (matches jax.nn.softmax).
// Raw v_exp_f32 / v_log_f32 (base-2) + v_rcp_f32: p+eps >= 1e-10 keeps us far from the
// denormal ranges the OCML wrappers guard against.
template <int N>
__device__ __forceinline__ float head_loss(const float* __restrict__ x,
                                           const float* __restrict__ lbl) {
    float m = x[0];
#pragma unroll
    for (int j = 1; j < N; ++j) m = fmaxf(m, x[j]);
    float e[N];
    float s = 0.0f;
#pragma unroll
    for (int j = 0; j < N; ++j) {
        e[j] = __builtin_amdgcn_exp2f((x[j] - m) * LOG2E);
        s += e[j];
    }
    const float inv = __builtin_amdgcn_rcpf(s);
    float acc = 0.0f;
#pragma unroll
    for (int j = 0; j < N; ++j)
        acc += __builtin_amdgcn_logf(e[j] * inv + LOSS_EPS) * lbl[j];  // log2
    return -LN2 * acc;
}

#define HAS_ASYNC_LDS __has_builtin(__builtin_amdgcn_global_load_async_to_lds_b128)

#if HAS_ASYNC_LDS
typedef __attribute__((address_space(1))) v4i glob_v4i;
typedef __attribute__((address_space(3))) v4i lds_v4i;
#endif

// Stage a contiguous run of float4s into LDS.
__device__ __forceinline__ void tile_copy_f4(float* __restrict__ dst,
                                             const float* __restrict__ src,
                                             int nfloat4, int tid) {
#if HAS_ASYNC_LDS
    const uintptr_t gbase = (uintptr_t)src;
    const uint32_t  lbase = (uint32_t)(uintptr_t)dst;  // low 32 bits = LDS byte offset
    for (int i = tid; i < nfloat4; i += 256) {
        __builtin_amdgcn_global_load_async_to_lds_b128(
            (glob_v4i*)(gbase + 16u * (uint32_t)i),
            (lds_v4i*)(lbase + 16u * (uint32_t)i),
            /*offset=*/0, /*cpol=*/0);
    }
#else
    const float4* g = (const float4*)src;
    float4*       d = (float4*)dst;
    for (int i = tid; i < nfloat4; i += 256) d[i] = g[i];
#endif
}

__device__ __forceinline__ void wait_tile_copies() {
#if HAS_ASYNC_LDS
#if __has_builtin(__builtin_amdgcn_s_wait_asynccnt)
    __builtin_amdgcn_s_wait_asynccnt(0);
#else
    asm volatile("s_wait_asynccnt 0x0" ::: "memory");
#endif
#endif
}

__global__ __launch_bounds__(256) void loss_partials_kernel(
    const float* __restrict__ o1, const float* __restrict__ o2,
    const float* __restrict__ o3, const float* __restrict__ o4,
    const float* __restrict__ lb, float* __restrict__ partials) {
    __shared__ float lds[TILE * 58];   // 59392 B staged tile
    __shared__ float wsum[8];

    float* s1 = lds;                 // TILE*8
    float* s2 = s1 + TILE * 8;       // TILE*6
    float* s3 = s2 + TILE * 6;       // TILE*6
    float* s4 = s3 + TILE * 6;       // TILE*9
    float* sl = s4 + TILE * 9;       // TILE*29

    const int t   = threadIdx.x;
    const int blk = blockIdx.x;

    tile_copy_f4(s1, o1 + (size_t)blk * TILE * 8,  TILE * 8  / 4, t);
    tile_copy_f4(s2, o2 + (size_t)blk * TILE * 6,  TILE * 6  / 4, t);
    tile_copy_f4(s3, o3 + (size_t)blk * TILE * 6,  TILE * 6  / 4, t);
    tile_copy_f4(s4, o4 + (size_t)blk * TILE * 9,  TILE * 9  / 4, t);
    tile_copy_f4(sl, lb + (size_t)blk * TILE * 29, TILE * 29 / 4, t);
    wait_tile_copies();
    __syncthreads();

    const float* lrow = sl + t * 29;
    float v = head_loss<8>(s1 + t * 8, lrow + 0)
            + head_loss<6>(s2 + t * 6, lrow + 8)
            + head_loss<6>(s3 + t * 6, lrow + 14)
            + head_loss<9>(s4 + t * 9, lrow + 20);

    const float ws = wave_sum_wmma(v);           // full EXEC here: all 256 threads live
    const int lane = t & 31, wid = t >> 5;
    if (lane == 0) wsum[wid] = ws;
    __syncthreads();
    if (t == 0) {
        float s = 0.0f;
#pragma unroll
        for (int i = 0; i < 8; ++i) s += wsum[i];
        partials[blk] = s;
    }
}

__global__ __launch_bounds__(256) void loss_finish_kernel(
    const float* __restrict__ partials, float* __restrict__ out) {
    __shared__ float wsum[8];
    const int t = threadIdx.x;
    float v = 0.0f;
#pragma unroll
    for (int i = 0; i < NBLK / 256; ++i) v += partials[t + i * 256];
    const float ws = wave_sum_wmma(v);
    if ((t & 31) == 0) wsum[t >> 5] = ws;
    __syncthreads();
    if (t == 0) {
        float s = 0.0f;
#pragma unroll
        for (int i = 0; i < 8; ++i) s += wsum[i];
        out[0] = s * (1.0f / (float)BTOT);       // exact power-of-2 scale
    }
}

extern "C" void kernel_launch(void* const* d_in, const int* in_sizes, int n_in,
                              void* d_out, int out_size, void* d_ws, size_t ws_size,
                              hipStream_t stream) {
    const float* o1 = (const float*)d_in[0];
    const float* o2 = (const float*)d_in[1];
    const float* o3 = (const float*)d_in[2];
    const float* o4 = (const float*)d_in[3];
    const float* lb = (const float*)d_in[4];
    float* partials = (float*)d_ws;              // 2048 floats, fully overwritten each call
    float* out      = (float*)d_out;

    loss_partials_kernel<<<NBLK, 256, 0, stream>>>(o1, o2, o3, o4, lb, partials);
    loss_finish_kernel<<<1, 256, 0, stream>>>(partials, out);
}